// HeteroGNN_85624468013339
// MI455X (gfx1250) — compile-verified
//
#include <hip/hip_runtime.h>
#include <hip/hip_bf16.h>

typedef __attribute__((ext_vector_type(2))) float v2f;
typedef __attribute__((ext_vector_type(8))) float v8f;

#define D_FEAT 32

// ---------------------------------------------------------------------------
// Utility kernels
// ---------------------------------------------------------------------------
__global__ void zero_f4_kernel(float4* __restrict__ p, int n4) {
    int i = blockIdx.x * blockDim.x + threadIdx.x;
    if (i < n4) {
        float4 z; z.x = 0.f; z.y = 0.f; z.z = 0.f; z.w = 0.f;
        p[i] = z;
    }
}

__global__ void count_deg_kernel(const int* __restrict__ idx, float* __restrict__ deg, int n) {
    int i = blockIdx.x * blockDim.x + threadIdx.x;
    if (i < n) atomicAdd(&deg[idx[i]], 1.0f);
}

__global__ void deg_to_scale_kernel(float* __restrict__ deg, int n) {
    int i = blockIdx.x * blockDim.x + threadIdx.x;
    if (i < n) {
        float d = deg[i];
        deg[i] = rsqrtf(fmaxf(d, 1.0f));
    }
}

// ---------------------------------------------------------------------------
// Edge scatter: one wave32 per edge, lane = feature index.
// agg[dst] += x[src] * s_out[src]   (L2-resident atomics on MI455X: agg is 64MB)
// ---------------------------------------------------------------------------
__global__ void scatter_edges_kernel(const float* __restrict__ x,
                                     const int* __restrict__ src,
                                     const int* __restrict__ dst,
                                     const float* __restrict__ s_out,
                                     float* __restrict__ agg,
                                     int e_count) {
    int lane = threadIdx.x & 31;
    int e = blockIdx.x * (blockDim.x >> 5) + (threadIdx.x >> 5);
    if (e >= e_count) return;
    int s = __builtin_amdgcn_readfirstlane(src[e]);  // wave-uniform
    int d = __builtin_amdgcn_readfirstlane(dst[e]);
    float val = x[(size_t)s * D_FEAT + lane] * s_out[s];
    atomicAdd(&agg[(size_t)d * D_FEAT + lane], val);
}

// ---------------------------------------------------------------------------
// WMMA helpers: fp32 16x16x4 WMMA, D = A(16x4) * B(4x16) + C.
// A layout (ISA 7.12.2, 32-bit A 16x4): lane m (0-15): K=4kb+{0,1};
// lane m+16: K=4kb+{2,3}.  B mirrors with rows striped across lanes.
// C/D: vgpr v, lanes 0-15 -> M=v, N=lane; lanes 16-31 -> M=v+8, N=lane-16.
// ---------------------------------------------------------------------------
__device__ __forceinline__ float elu1(float x) {
    return x > 0.0f ? x : (__expf(x) - 1.0f);
}

__device__ __forceinline__ void load_W_frags(const float* __restrict__ W,
                                             v2f bfrag[2][8], int lane) {
    int n = lane & 15;
    int kh = (lane >> 4) << 1;  // 0 or 2
#pragma unroll
    for (int nt = 0; nt < 2; ++nt) {
#pragma unroll
        for (int kb = 0; kb < 8; ++kb) {
            int k0 = 4 * kb + kh;
            bfrag[nt][kb].x = W[k0 * D_FEAT + 16 * nt + n];
            bfrag[nt][kb].y = W[(k0 + 1) * D_FEAT + 16 * nt + n];
        }
    }
}

__device__ __forceinline__ void load_A_frags(const float* __restrict__ X,
                                             const float* __restrict__ s_in,
                                             int row0, int lane, v2f afrag[8]) {
    int m = lane & 15;
    int kh = (lane >> 4) << 1;  // 0 or 2
    const float* xr = X + (size_t)(row0 + m) * D_FEAT;
    float sc = s_in[row0 + m];
#pragma unroll
    for (int kb = 0; kb < 8; ++kb) {
        afrag[kb].x = xr[4 * kb + kh] * sc;
        afrag[kb].y = xr[4 * kb + kh + 1] * sc;
    }
}

__device__ __forceinline__ v8f wmma_k32(const v2f a[8], const v2f b[8]) {
    v8f c = {0.f, 0.f, 0.f, 0.f, 0.f, 0.f, 0.f, 0.f};
#pragma unroll
    for (int kb = 0; kb < 8; ++kb) {
        c = __builtin_amdgcn_wmma_f32_16x16x4_f32(
            /*neg_a=*/false, a[kb], /*neg_b=*/false, b[kb],
            /*c_mod=*/(short)0, c, /*reuse_a=*/false, /*reuse_b=*/false);
    }
    return c;
}

// ---------------------------------------------------------------------------
// Finalize out_a: in-place  out = elu((agg * s_in) @ W + b)
// One wave per 16 rows. 16 WMMAs per wave.
// ---------------------------------------------------------------------------
__global__ void finalize_a_kernel(float* __restrict__ out,
                                  const float* __restrict__ s_in,
                                  const float* __restrict__ W,
                                  const float* __restrict__ bias, int n) {
    int lane = threadIdx.x & 31;
    int wave = threadIdx.x >> 5;
    int row0 = (blockIdx.x * (blockDim.x >> 5) + wave) * 16;
    if (row0 >= n) return;

    v2f bfrag[2][8];
    load_W_frags(W, bfrag, lane);
    v2f afrag[8];
    load_A_frags(out, s_in, row0, lane, afrag);

    v8f c0 = wmma_k32(afrag, bfrag[0]);
    v8f c1 = wmma_k32(afrag, bfrag[1]);

    int nn = lane & 15;
    float b0 = bias[nn];
    float b1 = bias[16 + nn];
    int mbase = (lane >> 4) << 3;
#pragma unroll
    for (int v = 0; v < 8; ++v) {
        float* orow = out + (size_t)(row0 + mbase + v) * D_FEAT;
        orow[nn]      = elu1(c0[v] + b0);
        orow[16 + nn] = elu1(c1[v] + b1);
    }
}

// ---------------------------------------------------------------------------
// Finalize out_b: out = elu((agg_ab*s_ab)@W + b) + elu((agg_bb*s_bb)@W + b)
// agg_ab lives in `out` (in place), agg_bb in workspace. 32 WMMAs per wave.
// ---------------------------------------------------------------------------
__global__ void finalize_b_kernel(float* __restrict__ out,
                                  const float* __restrict__ agg_bb,
                                  const float* __restrict__ s_ab,
                                  const float* __restrict__ s_bb,
                                  const float* __restrict__ W,
                                  const float* __restrict__ bias, int n) {
    int lane = threadIdx.x & 31;
    int wave = threadIdx.x >> 5;
    int row0 = (blockIdx.x * (blockDim.x >> 5) + wave) * 16;
    if (row0 >= n) return;

    v2f bfrag[2][8];
    load_W_frags(W, bfrag, lane);

    v2f a_ab[8];
    load_A_frags(out, s_ab, row0, lane, a_ab);
    v2f a_bb[8];
    load_A_frags(agg_bb, s_bb, row0, lane, a_bb);

    v8f c0 = wmma_k32(a_ab, bfrag[0]);
    v8f c1 = wmma_k32(a_ab, bfrag[1]);
    v8f c2 = wmma_k32(a_bb, bfrag[0]);
    v8f c3 = wmma_k32(a_bb, bfrag[1]);

    int nn = lane & 15;
    float b0 = bias[nn];
    float b1 = bias[16 + nn];
    int mbase = (lane >> 4) << 3;
#pragma unroll
    for (int v = 0; v < 8; ++v) {
        float* orow = out + (size_t)(row0 + mbase + v) * D_FEAT;
        orow[nn]      = elu1(c0[v] + b0) + elu1(c2[v] + b0);
        orow[16 + nn] = elu1(c1[v] + b1) + elu1(c3[v] + b1);
    }
}

// ---------------------------------------------------------------------------
// Launch
// ---------------------------------------------------------------------------
extern "C" void kernel_launch(void* const* d_in, const int* in_sizes, int n_in,
                              void* d_out, int out_size, void* d_ws, size_t ws_size,
                              hipStream_t stream) {
    const float* h_a  = (const float*)d_in[0];
    const float* h_b  = (const float*)d_in[1];
    const float* W    = (const float*)d_in[2];
    const float* bias = (const float*)d_in[3];
    const int* src_ab = (const int*)d_in[4];
    const int* dst_ab = (const int*)d_in[5];
    const int* src_bb = (const int*)d_in[6];
    const int* dst_bb = (const int*)d_in[7];
    const int* src_ba = (const int*)d_in[8];
    const int* dst_ba = (const int*)d_in[9];

    const int NA = 500000, NB = 500000, E = 2000000;

    float* out   = (float*)d_out;
    float* out_a = out;                        // doubles as agg_ba accumulator
    float* out_b = out + (size_t)NA * D_FEAT;  // doubles as agg_ab accumulator

    // Workspace layout (floats): agg_bb [NB*32] | 6 degree/scale arrays
    float* ws       = (float*)d_ws;
    float* agg_bb   = ws;
    float* degs     = ws + (size_t)NB * D_FEAT;
    float* s_out_ab = degs;                 // NA
    float* s_in_ab  = s_out_ab + NA;        // NB
    float* s_out_bb = s_in_ab + NB;         // NB
    float* s_in_bb  = s_out_bb + NB;        // NB
    float* s_out_ba = s_in_bb + NB;         // NB
    float* s_in_ba  = s_out_ba + NB;        // NA
    const int deg_total = 2 * NA + 4 * NB;  // 3,000,000

    const int T = 256;

    // 1) Zero output (agg accumulators) and workspace (agg_bb + degrees)
    {
        int n4 = ((NA + NB) * D_FEAT) / 4;  // 8M float4
        zero_f4_kernel<<<(n4 + T - 1) / T, T, 0, stream>>>((float4*)out, n4);
        int w4 = (NB * D_FEAT + deg_total) / 4;  // 4.75M float4
        zero_f4_kernel<<<(w4 + T - 1) / T, T, 0, stream>>>((float4*)ws, w4);
    }

    // 2) Degree counts (float atomic adds, exact for counts < 2^24)
    int eb = (E + T - 1) / T;
    count_deg_kernel<<<eb, T, 0, stream>>>(src_ab, s_out_ab, E);
    count_deg_kernel<<<eb, T, 0, stream>>>(dst_ab, s_in_ab, E);
    count_deg_kernel<<<eb, T, 0, stream>>>(src_bb, s_out_bb, E);
    count_deg_kernel<<<eb, T, 0, stream>>>(dst_bb, s_in_bb, E);
    count_deg_kernel<<<eb, T, 0, stream>>>(src_ba, s_out_ba, E);
    count_deg_kernel<<<eb, T, 0, stream>>>(dst_ba, s_in_ba, E);

    // 3) deg -> rsqrt(max(deg,1)) in place for all six arrays at once
    deg_to_scale_kernel<<<(deg_total + T - 1) / T, T, 0, stream>>>(degs, deg_total);

    // 4) Edge scatters (one wave per edge; 8 edges per 256-thread block)
    {
        int epb = T / 32;
        int gb = (E + epb - 1) / epb;
        scatter_edges_kernel<<<gb, T, 0, stream>>>(h_a, src_ab, dst_ab, s_out_ab, out_b, E);
        scatter_edges_kernel<<<gb, T, 0, stream>>>(h_b, src_bb, dst_bb, s_out_bb, agg_bb, E);
        scatter_edges_kernel<<<gb, T, 0, stream>>>(h_b, src_ba, dst_ba, s_out_ba, out_a, E);
    }

    // 5) WMMA finalize: 128 rows per 256-thread block (8 waves x 16 rows)
    {
        int rows_per_block = (T / 32) * 16;
        int ga = (NA + rows_per_block - 1) / rows_per_block;
        int gb2 = (NB + rows_per_block - 1) / rows_per_block;
        finalize_a_kernel<<<ga, T, 0, stream>>>(out_a, s_in_ba, W, bias, NA);
        finalize_b_kernel<<<gb2, T, 0, stream>>>(out_b, agg_bb, s_in_ab, s_in_bb, W, bias, NB);
    }
}